// ODEModel_78262894068397
// MI455X (gfx1250) — compile-verified
//
#include <hip/hip_runtime.h>
#include <math.h>

// Neural-ODE (dh/dt = W2 tanh(W1 h + b1) + b2) integrated with joint-batch
// adaptive dopri5 (jax.experimental.ode semantics), dense output at 100 times.
// Single persistent 1024-thread workgroup (32 wave32s) on one WGP:
//  - layer 1 via fp32 WMMA v_wmma_f32_16x16x4_f32 (K=2 padded to 4)
//  - tanh via hardware v_tanh_f32 (or branch-free exp2 fallback)
//  - tanh activations staged through LDS (intra-wave only -> no barrier)
//  - layer 2 scalar with ds_load_b128 vectorized reads
//  - block-wide error-norm reductions via shfl + LDS + s_barrier

#define BATCH 1024
#define HID   50
#define NT    100
#define RTOL  1e-7f
#define ATOL  1e-9f

typedef float v2f __attribute__((ext_vector_type(2)));
typedef float v8f __attribute__((ext_vector_type(8)));

#define U_STRIDE    52                      // row stride (floats), 16B-aligned
#define U_FLOATS    (BATCH * U_STRIDE)
#define W2_STRIDE   64                      // padded so both rows 16B-aligned
#define SMEM_FLOATS (U_FLOATS + 2 * W2_STRIDE + 64)

// ---------------- fast tanh: 1 trans op on gfx1250, else branch-free --------
__device__ __forceinline__ float fast_tanh(float x) {
#if __has_builtin(__builtin_amdgcn_tanhf)
  return __builtin_amdgcn_tanhf(x);
#elif __has_builtin(__builtin_amdgcn_exp2f)
  const float a = __builtin_fabsf(x);
  const float e = __builtin_amdgcn_exp2f(-2.885390081777927f * a); // exp(-2a)
  const float r = (1.0f - e) * __builtin_amdgcn_rcpf(1.0f + e);
  return __builtin_copysignf(r, x);
#else
  const float a = __builtin_fabsf(x);
  const float e = __expf(-2.0f * a);
  const float r = (1.0f - e) / (1.0f + e);
  return __builtin_copysignf(r, x);
#endif
}

// ---------------- block-wide sum over 1024 threads (32 waves) ----------------
__device__ __forceinline__ float blockSum(float v, float* red) {
  const int tid = threadIdx.x;
  #pragma unroll
  for (int o = 16; o > 0; o >>= 1) v += __shfl_xor(v, o, 32);
  __syncthreads();                       // protect `red` from previous use
  if ((tid & 31) == 0) red[tid >> 5] = v;
  __syncthreads();
  if (tid < 32) {
    float s = red[tid];
    #pragma unroll
    for (int o = 16; o > 0; o >>= 1) s += __shfl_xor(s, o, 32);
    if (tid == 0) red[32] = s;
  }
  __syncthreads();
  return red[32];
}

// ---------------- RHS: f = W2 tanh(W1 y + b1) + b2 ---------------------------
// Each wave owns batch rows [wave*32, wave*32+32). Layer 1 = 8 fp32 WMMAs
// (2 row-tiles x 4 col-tiles of 16), layer 2 = scalar via LDS-staged u.
__device__ __forceinline__ void rhs_eval(
    float y0, float y1, float& o0, float& o1,
    int lane, const v2f* __restrict__ bw, const float* __restrict__ b1v,
    float* __restrict__ uW,                  // this wave's 32 x U_STRIDE tile
    const float* __restrict__ w2S, float b20, float b21)
{
  const int  l16 = lane & 15;
  const bool lo  = lane < 16;
  #pragma unroll
  for (int T = 0; T < 2; ++T) {
    // Build A (16x4 f32, K=2 real): lanes 0-15 hold (M=l16, K=0/1),
    // lanes 16-31 are the K=2/3 slots -> zero.
    float ax = __shfl(y0, T * 16 + l16, 32);
    float ay = __shfl(y1, T * 16 + l16, 32);
    v2f A;
    A.x = lo ? ax : 0.0f;
    A.y = lo ? ay : 0.0f;
    #pragma unroll
    for (int c = 0; c < 4; ++c) {
      v8f acc = {};
      acc = __builtin_amdgcn_wmma_f32_16x16x4_f32(
          false, A, false, bw[c], (short)0, acc, false, false);
      const int   n  = c * 16 + l16;       // hidden-unit column this lane holds
      const float bb = b1v[c];
      float uv[8];
      #pragma unroll
      for (int r = 0; r < 8; ++r) uv[r] = fast_tanh(acc[r] + bb);
      if (n < HID) {                       // predicate only the stores
        const int rbase = T * 16 + (lo ? 0 : 8);
        #pragma unroll
        for (int r = 0; r < 8; ++r)
          uW[(rbase + r) * U_STRIDE + n] = uv[r];
      }
    }
  }
  // Layer 2: each lane reduces its own row (intra-wave LDS RAW, in-order DS).
  float s0 = b20, s1 = b21;
  const float*  urow = &uW[lane * U_STRIDE];
  const float4* u4   = (const float4*)urow;              // 16B-aligned
  const float4* wa4  = (const float4*)(w2S);             // w2 row 0
  const float4* wb4  = (const float4*)(w2S + W2_STRIDE); // w2 row 1
  #pragma unroll
  for (int q = 0; q < HID / 4; ++q) {      // 48 of 50
    const float4 u = u4[q], wa = wa4[q], wb = wb4[q];
    s0 = fmaf(u.x, wa.x, s0); s0 = fmaf(u.y, wa.y, s0);
    s0 = fmaf(u.z, wa.z, s0); s0 = fmaf(u.w, wa.w, s0);
    s1 = fmaf(u.x, wb.x, s1); s1 = fmaf(u.y, wb.y, s1);
    s1 = fmaf(u.z, wb.z, s1); s1 = fmaf(u.w, wb.w, s1);
  }
  const float u48 = urow[48], u49 = urow[49];
  s0 = fmaf(u48, w2S[48], s0);             s0 = fmaf(u49, w2S[49], s0);
  s1 = fmaf(u48, w2S[W2_STRIDE + 48], s1); s1 = fmaf(u49, w2S[W2_STRIDE + 49], s1);
  o0 = s0; o1 = s1;
}

#define RHS(yx, yy, ox, oy) rhs_eval((yx), (yy), (ox), (oy), lane, bw, b1v, uW, w2S, b20, b21)

__global__ __launch_bounds__(1024, 1) void node_dopri5_kernel(
    const float* __restrict__ h0,
    const float* __restrict__ tarr,
    const float* __restrict__ w1,
    const float* __restrict__ b1,
    const float* __restrict__ w2,
    const float* __restrict__ b2,
    float* __restrict__ out)
{
  extern __shared__ float smem[];
  float* uS  = smem;               // [BATCH][U_STRIDE] tanh activations
  float* w2S = smem + U_FLOATS;    // [2][W2_STRIDE]
  float* red = w2S + 2 * W2_STRIDE;

  const int tid  = threadIdx.x;
  const int lane = tid & 31;
  const int wave = tid >> 5;
  float* uW = &uS[wave * 32 * U_STRIDE];

  if (tid < 2 * HID) {
    const int d = tid / HID, n = tid - d * HID;
    w2S[d * W2_STRIDE + n] = w2[tid];      // w2 is [DIM][HID] row-major
  }
  __syncthreads();

  // Per-lane B tiles for layer-1 WMMA: B is 4x16 (K=2 real); lanes 0-15 hold
  // rows K=0 (x) / K=1 (y), lanes 16-31 are the K=2/3 rows -> zero.
  v2f bw[4]; float b1v[4];
  const int l16i = lane & 15;
  #pragma unroll
  for (int c = 0; c < 4; ++c) {
    const int  n = c * 16 + l16i;
    const bool v = (lane < 16) && (n < HID);
    bw[c].x = v ? w1[n * 2 + 0] : 0.0f;
    bw[c].y = v ? w1[n * 2 + 1] : 0.0f;
    b1v[c]  = (n < HID) ? b1[n] : 0.0f;
  }
  const float b20 = b2[0], b21 = b2[1];

  // ---- state: each thread owns one trajectory (DIM=2) ----
  float y0 = h0[tid * 2 + 0], y1 = h0[tid * 2 + 1];
  out[tid * 2 + 0] = y0;
  out[tid * 2 + 1] = y1;

  float f0, f1;
  RHS(y0, y1, f0, f1);

  // ---- select_initial_step (jax) ----
  const float invN = 1.0f / (2.0f * BATCH);
  const float sc0 = ATOL + fabsf(y0) * RTOL;
  const float sc1 = ATOL + fabsf(y1) * RTOL;
  float u0 = y0 / sc0, u1 = y1 / sc1;
  const float d0 = sqrtf(blockSum(u0 * u0 + u1 * u1, red) * invN);
  u0 = f0 / sc0; u1 = f1 / sc1;
  const float d1 = sqrtf(blockSum(u0 * u0 + u1 * u1, red) * invN);
  const float hh0 = (d0 < 1e-5f || d1 < 1e-5f) ? 1e-6f : 0.01f * d0 / d1;
  float p0, p1;
  RHS(y0 + hh0 * f0, y1 + hh0 * f1, p0, p1);
  u0 = (p0 - f0) / sc0; u1 = (p1 - f1) / sc1;
  const float d2 = sqrtf(blockSum(u0 * u0 + u1 * u1, red) * invN) / hh0;
  const float hh1 = (d1 <= 1e-15f && d2 <= 1e-15f)
                        ? fmaxf(1e-6f, hh0 * 1e-3f)
                        : __powf(0.01f / fmaxf(d1, d2), 0.2f);
  float dt = fminf(100.0f * hh0, hh1);

  // ---- dopri5 tableau ----
  const float A31 = (float)(3.0/40.0),      A32 = (float)(9.0/40.0);
  const float A41 = (float)(44.0/45.0),     A42 = (float)(-56.0/15.0),    A43 = (float)(32.0/9.0);
  const float A51 = (float)(19372.0/6561.0),A52 = (float)(-25360.0/2187.0),
              A53 = (float)(64448.0/6561.0),A54 = (float)(-212.0/729.0);
  const float A61 = (float)(9017.0/3168.0), A62 = (float)(-355.0/33.0),
              A63 = (float)(46732.0/5247.0),A64 = (float)(49.0/176.0),    A65 = (float)(-5103.0/18656.0);
  const float B1  = (float)(35.0/384.0),    B3  = (float)(500.0/1113.0),
              B4  = (float)(125.0/192.0),   B5  = (float)(-2187.0/6784.0),B6  = (float)(11.0/84.0);
  const float E1  = (float)(71.0/57600.0),  E3  = (float)(-71.0/16695.0), E4  = (float)(71.0/1920.0),
              E5  = (float)(-17253.0/339200.0), E6 = (float)(22.0/525.0), E7  = (float)(-1.0/40.0);
  const float CM1 = (float)(6025192743.0/30085553152.0/2.0),
              CM3 = (float)(51252292925.0/65400821598.0/2.0),
              CM4 = (float)(-2691868925.0/45128329728.0/2.0),
              CM5 = (float)(187940372067.0/1594534317056.0/2.0),
              CM6 = (float)(-1776094331.0/19743644256.0/2.0),
              CM7 = (float)(11237099.0/235043384.0/2.0);

  float t_now  = tarr[0];
  float t_prev = t_now;
  float Ac0 = 0, Bc0 = 0, Cc0 = 0, Dc0 = 0, Ec0 = y0;   // interp coeffs dim 0
  float Ac1 = 0, Bc1 = 0, Cc1 = 0, Dc1 = 0, Ec1 = y1;   // interp coeffs dim 1

  for (int i = 1; i < NT; ++i) {
    const float target = tarr[i];
    int guard = 0;
    while (t_now < target && guard++ < 100000) {
      // ---- one attempted dopri5 step of size dt (FSAL: k0 = f) ----
      const float k0x = f0, k0y = f1;
      float k1x, k1y, k2x, k2y, k3x, k3y, k4x, k4y, k5x, k5y, k6x, k6y;

      RHS(y0 + dt * (0.2f * k0x),
          y1 + dt * (0.2f * k0y), k1x, k1y);
      RHS(y0 + dt * (A31 * k0x + A32 * k1x),
          y1 + dt * (A31 * k0y + A32 * k1y), k2x, k2y);
      RHS(y0 + dt * (A41 * k0x + A42 * k1x + A43 * k2x),
          y1 + dt * (A41 * k0y + A42 * k1y + A43 * k2y), k3x, k3y);
      RHS(y0 + dt * (A51 * k0x + A52 * k1x + A53 * k2x + A54 * k3x),
          y1 + dt * (A51 * k0y + A52 * k1y + A53 * k2y + A54 * k3y), k4x, k4y);
      RHS(y0 + dt * (A61 * k0x + A62 * k1x + A63 * k2x + A64 * k3x + A65 * k4x),
          y1 + dt * (A61 * k0y + A62 * k1y + A63 * k2y + A64 * k3y + A65 * k4y), k5x, k5y);

      const float ynx = y0 + dt * (B1 * k0x + B3 * k2x + B4 * k3x + B5 * k4x + B6 * k5x);
      const float yny = y1 + dt * (B1 * k0y + B3 * k2y + B4 * k3y + B5 * k4y + B6 * k5y);
      RHS(ynx, yny, k6x, k6y);

      const float ex = dt * (E1 * k0x + E3 * k2x + E4 * k3x + E5 * k4x + E6 * k5x + E7 * k6x);
      const float ey = dt * (E1 * k0y + E3 * k2y + E4 * k3y + E5 * k4y + E6 * k5y + E7 * k6y);

      const float tol0 = ATOL + RTOL * fmaxf(fabsf(y0), fabsf(ynx));
      const float tol1 = ATOL + RTOL * fmaxf(fabsf(y1), fabsf(yny));
      const float r0 = ex / tol0, r1 = ey / tol1;
      const float ratio = sqrtf(blockSum(r0 * r0 + r1 * r1, red) * invN);

      if (ratio <= 1.0f) {
        // accepted: fit 4th-order interpolant over [t_now, t_now + dt]
        const float mx = y0 + dt * (CM1 * k0x + CM3 * k2x + CM4 * k3x + CM5 * k4x + CM6 * k5x + CM7 * k6x);
        const float my = y1 + dt * (CM1 * k0y + CM3 * k2y + CM4 * k3y + CM5 * k4y + CM6 * k5y + CM7 * k6y);
        const float g0x = dt * k0x, g1x = dt * k6x;
        const float g0y = dt * k0y, g1y = dt * k6y;
        Ac0 = -2.f*g0x + 2.f*g1x -  8.f*y0 -  8.f*ynx + 16.f*mx;
        Bc0 =  5.f*g0x - 3.f*g1x + 18.f*y0 + 14.f*ynx - 32.f*mx;
        Cc0 = -4.f*g0x +     g1x - 11.f*y0 -  5.f*ynx + 16.f*mx;
        Dc0 = g0x; Ec0 = y0;
        Ac1 = -2.f*g0y + 2.f*g1y -  8.f*y1 -  8.f*yny + 16.f*my;
        Bc1 =  5.f*g0y - 3.f*g1y + 18.f*y1 + 14.f*yny - 32.f*my;
        Cc1 = -4.f*g0y +     g1y - 11.f*y1 -  5.f*yny + 16.f*my;
        Dc1 = g0y; Ec1 = y1;
        y0 = ynx; y1 = yny; f0 = k6x; f1 = k6y;     // FSAL
        t_prev = t_now; t_now += dt;
      }
      // jax optimal_step_size (safety .9, ifactor 10, dfactor .2, order 5)
      const float dlow   = (ratio < 1.0f) ? 1.0f : 0.2f;
      const float factor = fminf(10.0f, fmaxf(0.9f * __powf(ratio, -0.2f), dlow));
      dt = (ratio == 0.0f) ? dt * 10.0f : dt * factor;
    }
    // dense output at target via last accepted step's interpolant
    const float span = t_now - t_prev;
    const float x    = (span > 0.0f) ? (target - t_prev) / span : 1.0f;
    out[i * (2 * BATCH) + tid * 2 + 0] =
        (((Ac0 * x + Bc0) * x + Cc0) * x + Dc0) * x + Ec0;
    out[i * (2 * BATCH) + tid * 2 + 1] =
        (((Ac1 * x + Bc1) * x + Cc1) * x + Dc1) * x + Ec1;
  }
}

extern "C" void kernel_launch(void* const* d_in, const int* in_sizes, int n_in,
                              void* d_out, int out_size, void* d_ws, size_t ws_size,
                              hipStream_t stream) {
  (void)in_sizes; (void)n_in; (void)out_size; (void)d_ws; (void)ws_size;
  const float* h0 = (const float*)d_in[0];
  const float* t  = (const float*)d_in[1];
  const float* w1 = (const float*)d_in[2];
  const float* b1 = (const float*)d_in[3];
  const float* w2 = (const float*)d_in[4];
  const float* b2 = (const float*)d_in[5];
  float* out = (float*)d_out;
  const size_t shmem = (size_t)SMEM_FLOATS * sizeof(float);   // ~209 KB < 320 KB/WGP
  node_dopri5_kernel<<<1, BATCH, shmem, stream>>>(h0, t, w1, b1, w2, b2, out);
}